// SerialIntervalGraph_32830730010781
// MI455X (gfx1250) — compile-verified
//
#include <hip/hip_runtime.h>

typedef __attribute__((ext_vector_type(2))) float v2f;
typedef __attribute__((ext_vector_type(8))) float v8f;

#define B_    16
#define T_    64
#define N_    2048
#define H_    32
#define D_    16
#define LAG_  7
#define ASTRIDE 2052   // 2048 + 4: skews LDS banks by 4*row -> conflict-free column reads

// ---------------------------------------------------------------------------
// Kernel 1: QKV projections.  features[B,N,32] x {Wq[32,16],Wk[32,16],Wv[32,32]}
// One thread per output element (64 outputs per (b,n) row).
// ---------------------------------------------------------------------------
__global__ __launch_bounds__(256) void sig_qkv_kernel(
    const float* __restrict__ features,
    const float* __restrict__ Wk, const float* __restrict__ bk,
    const float* __restrict__ Wq, const float* __restrict__ bq,
    const float* __restrict__ Wv, const float* __restrict__ bv,
    float* __restrict__ Qb, float* __restrict__ Kb, float* __restrict__ Vb)
{
    int g = blockIdx.x * blockDim.x + threadIdx.x;
    const int total = B_ * N_ * 64;
    if (g >= total) return;
    int o  = g & 63;
    int bn = g >> 6;                       // b*N + n
    const float* f = features + bn * H_;
    if (o < 16) {                          // queries
        float acc = bq[o];
        #pragma unroll
        for (int k = 0; k < 32; ++k) acc += f[k] * Wq[k * 16 + o];
        Qb[bn * 16 + o] = acc;
    } else if (o < 32) {                   // keys
        int d = o - 16;
        float acc = bk[d];
        #pragma unroll
        for (int k = 0; k < 32; ++k) acc += f[k] * Wk[k * 16 + d];
        Kb[bn * 16 + d] = acc;
    } else {                               // values
        int h = o - 32;
        float acc = bv[h];
        #pragma unroll
        for (int k = 0; k < 32; ++k) acc += f[k] * Wv[k * 32 + h];
        Vb[bn * 32 + h] = acc;
    }
}

// ---------------------------------------------------------------------------
// Kernel 2: fused affinity -> softmax -> P@V -> +temporal -> Wo -> LayerNorm.
// One block = (batch b, 16-row tile).  256 threads = 8 wave32.
// Dynamic LDS layout (floats):
//   [0 .. 16*ASTRIDE)                : aff block (16 x 2048, padded stride)
//                                      (reused later as 8x512 wave partials)
//   [16*ASTRIDE .. +16)              : weighted temporal term per row
//   next 512                         : out_pre (prop + temporal)
//   next 512                         : out2 (after Wo)
// ---------------------------------------------------------------------------
__global__ __launch_bounds__(256) void sig_main_kernel(
    const float* __restrict__ x,
    const float* __restrict__ adj_prior,
    const float* __restrict__ delay_w,
    const float* __restrict__ adj_weight,
    const float* __restrict__ Qb, const float* __restrict__ Kb,
    const float* __restrict__ Vb,
    const float* __restrict__ Wo, const float* __restrict__ bo,
    const float* __restrict__ ln_g, const float* __restrict__ ln_b,
    float* __restrict__ out)
{
    extern __shared__ float smem[];
    float* aff  = smem;                         // 16 * ASTRIDE
    float* wgt  = smem + 16 * ASTRIDE;          // 16
    float* opre = wgt + 16;                     // 512
    float* o2   = opre + 512;                   // 512

    const int b    = blockIdx.x >> 7;           // / 128
    const int n0   = (blockIdx.x & 127) << 4;   // row-tile start
    const int t    = threadIdx.x;
    const int wave = t >> 5;
    const int lane = t & 31;
    const int half = lane >> 4;                 // 0: lanes 0-15, 1: lanes 16-31
    const int ml   = lane & 15;                 // M (or N) index within tile

    const float blend = 1.0f / (1.0f + __expf(-adj_weight[0]));

    // --- temporal delay-weighted term: weighted[r] = sum_tau dw[tau]*x[b,63-tau,n0+r]
    if (t < 16) {
        float dmax = delay_w[0];
        #pragma unroll
        for (int i = 1; i < LAG_; ++i) dmax = fmaxf(dmax, delay_w[i]);
        float e[LAG_], ds = 0.0f;
        #pragma unroll
        for (int i = 0; i < LAG_; ++i) { e[i] = __expf(delay_w[i] - dmax); ds += e[i]; }
        float wv = 0.0f;
        #pragma unroll
        for (int i = 0; i < LAG_; ++i)
            wv += (e[i] / ds) * x[((size_t)b * T_ + (T_ - 1 - i)) * N_ + n0 + t];
        wgt[t] = wv;
    }

    // --- load Q tile A-fragments (16x16, as 4 K-steps of 16x4)
    // A 16x4 layout: lane ml holds M=ml; VGPR0 = K = 2*half, VGPR1 = K = 2*half+1
    v2f afr[4];
    {
        const float* q = Qb + ((size_t)b * N_ + n0 + ml) * D_;
        #pragma unroll
        for (int kk = 0; kk < 4; ++kk) {
            int base = kk * 4 + half * 2;
            v2f av = { q[base], q[base + 1] };
            afr[kk] = av;
        }
    }

    // ================= Phase A: aff row-block [16 x 2048] into LDS ============
    const float sc = 0.25f * (1.0f - blend);    // 1/sqrt(D) * (1-blend)
    for (int j = 0; j < 16; ++j) {
        const int gc = wave * 256 + j * 16 + ml;        // affinity column
        const float* kp = Kb + ((size_t)b * N_ + gc) * D_;
        v8f acc = {0.f,0.f,0.f,0.f,0.f,0.f,0.f,0.f};
        #pragma unroll
        for (int kk = 0; kk < 4; ++kk) {
            int base = kk * 4 + half * 2;
            v2f bv2 = { kp[base], kp[base + 1] };       // B[k][col] = keys[col][k]
            acc = __builtin_amdgcn_wmma_f32_16x16x4_f32(
                false, afr[kk], false, bv2, (short)0, acc, false, false);
        }
        #pragma unroll
        for (int v = 0; v < 8; ++v) {
            int row = v + half * 8;                     // C layout: VGPR v -> M=v(+8)
            aff[row * ASTRIDE + gc] =
                sc * acc[v] + blend * adj_prior[(size_t)(n0 + row) * N_ + gc];
        }
    }
    __syncthreads();

    // ================= Phase B: softmax per row (16 threads / row) ============
    {
        const int r = t >> 4, sub = t & 15;
        float* rowp = aff + r * ASTRIDE;
        float m = -3.402823466e38f;
        for (int i = 0; i < 128; ++i) m = fmaxf(m, rowp[sub + 16 * i]);
        #pragma unroll
        for (int s = 1; s < 16; s <<= 1) m = fmaxf(m, __shfl_xor(m, s, 32));
        float sum = 0.0f;
        for (int i = 0; i < 128; ++i) {
            float e = __expf(rowp[sub + 16 * i] - m);
            rowp[sub + 16 * i] = e;
            sum += e;
        }
        #pragma unroll
        for (int s = 1; s < 16; s <<= 1) sum += __shfl_xor(sum, s, 32);
        float inv = 1.0f / sum;
        for (int i = 0; i < 128; ++i) rowp[sub + 16 * i] *= inv;
    }
    __syncthreads();

    // ================= Phase C: propagated = P @ V  (16 x 32) ================
    v8f c0 = {0.f,0.f,0.f,0.f,0.f,0.f,0.f,0.f};
    v8f c1 = {0.f,0.f,0.f,0.f,0.f,0.f,0.f,0.f};
    for (int chunk = 0; chunk < 16; ++chunk) {
        const int kt = wave * 256 + chunk * 16;
        #pragma unroll
        for (int kk = 0; kk < 4; ++kk) {
            const int kidx = kt + kk * 4 + half * 2;
            // A fragment from P (LDS): row=ml, K = kidx, kidx+1
            v2f pa = { aff[ml * ASTRIDE + kidx], aff[ml * ASTRIDE + kidx + 1] };
            // B fragments from V: B[k][h], h = ml (tile0) / ml+16 (tile1)
            const float* v0 = Vb + ((size_t)b * N_ + kidx) * H_;
            const float* v1 = Vb + ((size_t)b * N_ + kidx + 1) * H_;
            v2f b0 = { v0[ml],      v1[ml]      };
            v2f b1 = { v0[ml + 16], v1[ml + 16] };
            c0 = __builtin_amdgcn_wmma_f32_16x16x4_f32(
                false, pa, false, b0, (short)0, c0, false, false);
            c1 = __builtin_amdgcn_wmma_f32_16x16x4_f32(
                false, pa, false, b1, (short)0, c1, false, false);
        }
    }
    __syncthreads();            // all P reads done -> may overwrite aff region

    // --- cross-wave reduction of the 8 partial [16x32] accumulators ----------
    float* part = smem;         // reuse aff LDS: 8 waves * 512 floats
    #pragma unroll
    for (int v = 0; v < 8; ++v) {
        int row = v + half * 8;
        part[wave * 512 + row * 32 + ml]      = c0[v];
        part[wave * 512 + row * 32 + ml + 16] = c1[v];
    }
    __syncthreads();

    for (int e = t; e < 512; e += 256) {
        float s = 0.0f;
        #pragma unroll
        for (int w2 = 0; w2 < 8; ++w2) s += part[w2 * 512 + e];
        opre[e] = s + 0.1f * wgt[e >> 5];
    }
    __syncthreads();

    // --- Wo projection: out2 = out_pre @ Wo + bo -----------------------------
    for (int e = t; e < 512; e += 256) {
        const int row = e >> 5, h = e & 31;
        const float* rp = opre + row * 32;
        float s = bo[h];
        #pragma unroll
        for (int k = 0; k < 32; ++k) s += rp[k] * Wo[k * 32 + h];
        o2[e] = s;
    }
    __syncthreads();

    // --- LayerNorm over H=32 (one wave handles 2 rows, lane = h) -------------
    #pragma unroll
    for (int rr = 0; rr < 2; ++rr) {
        const int row = wave * 2 + rr;
        float vv = o2[row * 32 + lane];
        float mu = vv;
        #pragma unroll
        for (int s = 1; s < 32; s <<= 1) mu += __shfl_xor(mu, s, 32);
        mu *= (1.0f / 32.0f);
        float d = vv - mu;
        float var = d * d;
        #pragma unroll
        for (int s = 1; s < 32; s <<= 1) var += __shfl_xor(var, s, 32);
        var *= (1.0f / 32.0f);
        out[((size_t)b * N_ + n0 + row) * H_ + lane] =
            d * rsqrtf(var + 1e-5f) * ln_g[lane] + ln_b[lane];
    }
}

// ---------------------------------------------------------------------------
extern "C" void kernel_launch(void* const* d_in, const int* in_sizes, int n_in,
                              void* d_out, int out_size, void* d_ws, size_t ws_size,
                              hipStream_t stream) {
    const float* x          = (const float*)d_in[0];
    const float* features   = (const float*)d_in[1];
    const float* adj_prior  = (const float*)d_in[2];
    const float* delay_w    = (const float*)d_in[3];
    const float* adj_weight = (const float*)d_in[4];
    const float* Wk = (const float*)d_in[5];
    const float* bk = (const float*)d_in[6];
    const float* Wq = (const float*)d_in[7];
    const float* bq = (const float*)d_in[8];
    const float* Wv = (const float*)d_in[9];
    const float* bv = (const float*)d_in[10];
    const float* Wo = (const float*)d_in[11];
    const float* bo = (const float*)d_in[12];
    const float* ln_g = (const float*)d_in[13];
    const float* ln_b = (const float*)d_in[14];
    float* out = (float*)d_out;

    float* Qb = (float*)d_ws;                 // [B,N,16]
    float* Kb = Qb + (size_t)B_ * N_ * D_;    // [B,N,16]
    float* Vb = Kb + (size_t)B_ * N_ * D_;    // [B,N,32]

    // QKV projections
    {
        const int total = B_ * N_ * 64;
        sig_qkv_kernel<<<(total + 255) / 256, 256, 0, stream>>>(
            features, Wk, bk, Wq, bq, Wv, bv, Qb, Kb, Vb);
    }

    // Fused attention + epilogue
    {
        const size_t smem_bytes = (size_t)(16 * ASTRIDE + 16 + 512 + 512) * sizeof(float);
        sig_main_kernel<<<B_ * (N_ / 16), 256, smem_bytes, stream>>>(
            x, adj_prior, delay_w, adj_weight, Qb, Kb, Vb,
            Wo, bo, ln_g, ln_b, out);
    }
}